// ResGatedGraphConv__61237643706857
// MI455X (gfx1250) — compile-verified
//
#include <hip/hip_runtime.h>
#include <hip/hip_bf16.h>

typedef __attribute__((ext_vector_type(2))) float v2f;
typedef __attribute__((ext_vector_type(8))) float v8f;

#define NDIM 256
#define H1   5
#define NGRP 64
#define EPSN 1e-5f

__device__ __forceinline__ float sigm(float x) {
    return 1.0f / (1.0f + __expf(-x));
}

// ---------------------------------------------------------------------------
// 0) zero the small group-stat scratch (gsum 320 + gcnt 64 + vsum 320)
// ---------------------------------------------------------------------------
__global__ void k_init(float* gsum, float* gcnt, float* vsum) {
    int t = threadIdx.x;
    if (t < NGRP * H1) { gsum[t] = 0.f; vsum[t] = 0.f; }
    if (t < NGRP) gcnt[t] = 0.f;
}

// ---------------------------------------------------------------------------
// 1) pack Wcat[256][32] = [Wk1|Wq1|Wv1|Ws1|pad]; biasvec[32]=[bk1,bq1,bv1,b1]
// ---------------------------------------------------------------------------
__global__ void k_pack(const float* __restrict__ Wk1, const float* __restrict__ Wq1,
                       const float* __restrict__ Wv1, const float* __restrict__ Ws1,
                       const float* __restrict__ bk1, const float* __restrict__ bq1,
                       const float* __restrict__ bv1, const float* __restrict__ b1,
                       float* __restrict__ Wcat, float* __restrict__ biasvec) {
    int t = blockIdx.x * blockDim.x + threadIdx.x;
    if (t < NDIM * 32) {
        int k = t >> 5, c = t & 31;
        float v = 0.f;
        if      (c < 5)  v = Wk1[k * H1 + c];
        else if (c < 10) v = Wq1[k * H1 + c - 5];
        else if (c < 15) v = Wv1[k * H1 + c - 10];
        else if (c < 20) v = Ws1[k * H1 + c - 15];
        Wcat[k * 32 + c] = v;
    }
    if (t < 32) {
        float v = 0.f;
        if      (t < 5)  v = bk1[t];
        else if (t < 10) v = bq1[t - 5];
        else if (t < 15) v = bv1[t - 10];
        else if (t < 20) v = b1[t - 15];
        biasvec[t] = v;
    }
}

// ---------------------------------------------------------------------------
// 2) single pass over edge_attr: e1 = ea@We1+be1 (5), e2 = ea@We2+be2 (1)
//    eproj layout: 6 floats per edge
// ---------------------------------------------------------------------------
__global__ __launch_bounds__(256) void k_edge_proj(
    const float* __restrict__ ea,
    const float* __restrict__ We1, const float* __restrict__ be1,
    const float* __restrict__ We2, const float* __restrict__ be2,
    float* __restrict__ eproj, int E_) {
    __shared__ float sW[32 * 6];   // [k][0..4]=We1 row, [k][5]=We2
    for (int i = threadIdx.x; i < 32 * 6; i += 256) {
        int k = i / 6, j = i % 6;
        sW[i] = (j < 5) ? We1[k * H1 + j] : We2[k];
    }
    __syncthreads();
    int e = blockIdx.x * blockDim.x + threadIdx.x;
    if (e >= E_) return;
    const float* row = ea + (size_t)e * 32;
    float acc[6] = { be1[0], be1[1], be1[2], be1[3], be1[4], be2[0] };
    #pragma unroll 4
    for (int k = 0; k < 32; ++k) {
        float v = row[k];
        const float* w = &sW[k * 6];
        acc[0] += v * w[0]; acc[1] += v * w[1]; acc[2] += v * w[2];
        acc[3] += v * w[3]; acc[4] += v * w[4]; acc[5] += v * w[5];
    }
    float* out = eproj + (size_t)e * 6;
    out[0] = acc[0]; out[1] = acc[1]; out[2] = acc[2];
    out[3] = acc[3]; out[4] = acc[4]; out[5] = acc[5];
}

// ---------------------------------------------------------------------------
// 3) WMMA node GEMM: [N x 256] @ [256 x 32] (padded). Per wave: one 16-row
//    tile, two 16-col N-tiles, 64 K-steps of V_WMMA_F32_16X16X4_F32.
//    Outputs: kqv[N][16] (cols 0..14 = k,q,v) ; agg1[N][5] = x@Ws1 + b1.
// ---------------------------------------------------------------------------
__global__ __launch_bounds__(256) void k_node_gemm(
    const float* __restrict__ x, const float* __restrict__ Wcat,
    const float* __restrict__ biasvec,
    float* __restrict__ kqv, float* __restrict__ agg1, int N_) {
    __shared__ float sW[NDIM * 32];          // 32 KB: whole weight block
    for (int i = threadIdx.x; i < NDIM * 32; i += 256) sW[i] = Wcat[i];
    __syncthreads();

    int lane = threadIdx.x & 31;
    int wave = threadIdx.x >> 5;
    int tile = blockIdx.x * 8 + wave;        // 16-row M tile
    int m0   = tile * 16;
    if (m0 >= N_) return;                    // whole wave exits together

    int hl = lane & 15;                      // half-lane index
    int hi = lane >> 4;                      // 0: K+0/1, 1: K+2/3
    int arow = m0 + hl; if (arow >= N_) arow = N_ - 1;
    const float* xrow = x + (size_t)arow * NDIM;

    v8f acc0 = {}; v8f acc1 = {};
    #pragma unroll 8
    for (int kk = 0; kk < NDIM; kk += 4) {
        int kb = kk + 2 * hi;
        // A fragment: A[M=hl][K=kb], A[M=hl][K=kb+1]
        float2 av = *(const float2*)(xrow + kb);
        v2f a; a.x = av.x; a.y = av.y;
        // B fragments: B[K=kb..kb+1][N = hl (+16)]
        v2f b0, b1;
        b0.x = sW[kb * 32 + hl];       b0.y = sW[(kb + 1) * 32 + hl];
        b1.x = sW[kb * 32 + 16 + hl];  b1.y = sW[(kb + 1) * 32 + 16 + hl];
        acc0 = __builtin_amdgcn_wmma_f32_16x16x4_f32(false, a, false, b0,
                                                     (short)0, acc0, false, false);
        acc1 = __builtin_amdgcn_wmma_f32_16x16x4_f32(false, a, false, b1,
                                                     (short)0, acc1, false, false);
    }

    // C layout: acc[r] -> row m0 + r + 8*hi, col hl (tile0) / 16+hl (tile1)
    #pragma unroll
    for (int r = 0; r < 8; ++r) {
        int row = m0 + r + 8 * hi;
        if (row >= N_) continue;
        int c0 = hl;
        float v0 = acc0[r] + biasvec[c0];
        if (c0 < 15)       kqv[(size_t)row * 16 + c0] = v0;
        else               agg1[(size_t)row * H1 + 0] = v0;     // c0 == 15 -> s0
        int c1 = 16 + hl;
        if (c1 < 20) {
            float v1 = acc1[r] + biasvec[c1];
            agg1[(size_t)row * H1 + (c1 - 15)] = v1;            // s1..s4
        }
    }
}

// ---------------------------------------------------------------------------
// 4) conv1 edge pass: gate = sigmoid(k[d]+q[s]+2e) ; atomic add into agg1[d]
// ---------------------------------------------------------------------------
__global__ __launch_bounds__(256) void k_edge_conv1(
    const int* __restrict__ src, const int* __restrict__ dst,
    const float* __restrict__ kqv, const float* __restrict__ eproj,
    float* __restrict__ agg1, int E_) {
    int e = blockIdx.x * blockDim.x + threadIdx.x;
    if (e >= E_) return;
    int s = src[e], d = dst[e];
    const float* kd = kqv + (size_t)d * 16;        // k of dst
    const float* qs = kqv + (size_t)s * 16 + 5;    // q of src
    const float* vs = kqv + (size_t)s * 16 + 10;   // v of src
    const float* ep = eproj + (size_t)e * 6;
    #pragma unroll
    for (int h = 0; h < H1; ++h) {
        float eh = ep[h];
        float g  = sigm(kd[h] + qs[h] + 2.0f * eh);
        atomicAdd(&agg1[(size_t)d * H1 + h], g * (vs[h] + eh));
    }
}

// ---------------------------------------------------------------------------
// 5) GraphNorm pass 1: per-group count + feature sums (LDS then global atomics)
// ---------------------------------------------------------------------------
__global__ __launch_bounds__(256) void k_gn_sum(
    const float* __restrict__ agg1, const int* __restrict__ batch,
    float* __restrict__ gsum, float* __restrict__ gcnt, int N_) {
    __shared__ float ls[NGRP * H1];
    __shared__ float lc[NGRP];
    for (int i = threadIdx.x; i < NGRP * H1; i += 256) ls[i] = 0.f;
    for (int i = threadIdx.x; i < NGRP; i += 256) lc[i] = 0.f;
    __syncthreads();
    int n = blockIdx.x * 256 + threadIdx.x;
    if (n < N_) {
        int g = batch[n];
        atomicAdd(&lc[g], 1.f);
        #pragma unroll
        for (int h = 0; h < H1; ++h)
            atomicAdd(&ls[g * H1 + h], agg1[(size_t)n * H1 + h]);
    }
    __syncthreads();
    for (int i = threadIdx.x; i < NGRP * H1; i += 256)
        if (ls[i] != 0.f) atomicAdd(&gsum[i], ls[i]);
    for (int i = threadIdx.x; i < NGRP; i += 256)
        if (lc[i] != 0.f) atomicAdd(&gcnt[i], lc[i]);
}

__global__ void k_gn_mean(const float* __restrict__ gsum,
                          const float* __restrict__ gcnt,
                          float* __restrict__ mean) {
    int t = threadIdx.x;
    if (t < NGRP * H1) {
        float c = gcnt[t / H1];
        mean[t] = gsum[t] / fmaxf(c, 1.0f);
    }
}

// 5b) center + accumulate variance sums
__global__ __launch_bounds__(256) void k_gn_center(
    const float* __restrict__ agg1, const int* __restrict__ batch,
    const float* __restrict__ mean, const float* __restrict__ gms,
    float* __restrict__ hbuf, float* __restrict__ vsum, int N_) {
    __shared__ float ls[NGRP * H1];
    for (int i = threadIdx.x; i < NGRP * H1; i += 256) ls[i] = 0.f;
    __syncthreads();
    int n = blockIdx.x * 256 + threadIdx.x;
    if (n < N_) {
        int g = batch[n];
        #pragma unroll
        for (int h = 0; h < H1; ++h) {
            float o = agg1[(size_t)n * H1 + h] - mean[g * H1 + h] * gms[h];
            hbuf[(size_t)n * H1 + h] = o;
            atomicAdd(&ls[g * H1 + h], o * o);
        }
    }
    __syncthreads();
    for (int i = threadIdx.x; i < NGRP * H1; i += 256)
        if (ls[i] != 0.f) atomicAdd(&vsum[i], ls[i]);
}

__global__ void k_gn_istd(const float* __restrict__ vsum,
                          const float* __restrict__ gcnt,
                          float* __restrict__ istd) {
    int t = threadIdx.x;
    if (t < NGRP * H1) {
        float c = fmaxf(gcnt[t / H1], 1.0f);
        istd[t] = rsqrtf(vsum[t] / c + EPSN);
    }
}

// ---------------------------------------------------------------------------
// 6) normalize + ReLU, fused with conv2 node projections (H2 == 1)
//    node3[n] = {k2, q2, v2};  agg2[n] = h@Ws2 + b2 (root term)
// ---------------------------------------------------------------------------
__global__ __launch_bounds__(256) void k_gn_final(
    const float* __restrict__ hbuf, const int* __restrict__ batch,
    const float* __restrict__ istd,
    const float* __restrict__ gw, const float* __restrict__ gb,
    const float* __restrict__ Wk2, const float* __restrict__ bk2,
    const float* __restrict__ Wq2, const float* __restrict__ bq2,
    const float* __restrict__ Wv2, const float* __restrict__ bv2,
    const float* __restrict__ Ws2, const float* __restrict__ b2,
    float* __restrict__ node3, float* __restrict__ agg2, int N_) {
    int n = blockIdx.x * blockDim.x + threadIdx.x;
    if (n >= N_) return;
    int g = batch[n];
    float k2 = bk2[0], q2 = bq2[0], v2 = bv2[0], s2 = b2[0];
    #pragma unroll
    for (int h = 0; h < H1; ++h) {
        float y = gw[h] * hbuf[(size_t)n * H1 + h] * istd[g * H1 + h] + gb[h];
        y = fmaxf(y, 0.f);                                    // ReLU
        k2 += y * Wk2[h]; q2 += y * Wq2[h];
        v2 += y * Wv2[h]; s2 += y * Ws2[h];
    }
    float* o = node3 + (size_t)n * 4;
    o[0] = k2; o[1] = q2; o[2] = v2;
    agg2[n] = s2;
}

// ---------------------------------------------------------------------------
// 7) conv2 edge pass (scalar) + final sigmoid
// ---------------------------------------------------------------------------
__global__ __launch_bounds__(256) void k_edge_conv2(
    const int* __restrict__ src, const int* __restrict__ dst,
    const float* __restrict__ node3, const float* __restrict__ eproj,
    float* __restrict__ agg2, int E_) {
    int e = blockIdx.x * blockDim.x + threadIdx.x;
    if (e >= E_) return;
    int s = src[e], d = dst[e];
    float e2 = eproj[(size_t)e * 6 + 5];
    float g  = sigm(node3[(size_t)d * 4 + 0] + node3[(size_t)s * 4 + 1] + 2.0f * e2);
    atomicAdd(&agg2[d], g * (node3[(size_t)s * 4 + 2] + e2));
}

__global__ void k_final(const float* __restrict__ agg2, float* __restrict__ out, int N_) {
    int n = blockIdx.x * blockDim.x + threadIdx.x;
    if (n < N_) out[n] = sigm(agg2[n]);
}

// ---------------------------------------------------------------------------
extern "C" void kernel_launch(void* const* d_in, const int* in_sizes, int n_in,
                              void* d_out, int out_size, void* d_ws, size_t ws_size,
                              hipStream_t stream) {
    const float* x    = (const float*)d_in[0];
    const float* ea   = (const float*)d_in[1];
    const int*   eidx = (const int*)d_in[2];
    const int*   batch= (const int*)d_in[3];
    const float* Wk1 = (const float*)d_in[4];  const float* bk1 = (const float*)d_in[5];
    const float* Wq1 = (const float*)d_in[6];  const float* bq1 = (const float*)d_in[7];
    const float* Wv1 = (const float*)d_in[8];  const float* bv1 = (const float*)d_in[9];
    const float* We1 = (const float*)d_in[10]; const float* be1 = (const float*)d_in[11];
    const float* Ws1 = (const float*)d_in[12]; const float* b1  = (const float*)d_in[13];
    const float* gw  = (const float*)d_in[14]; const float* gb  = (const float*)d_in[15];
    const float* gms = (const float*)d_in[16];
    const float* Wk2 = (const float*)d_in[17]; const float* bk2 = (const float*)d_in[18];
    const float* Wq2 = (const float*)d_in[19]; const float* bq2 = (const float*)d_in[20];
    const float* Wv2 = (const float*)d_in[21]; const float* bv2 = (const float*)d_in[22];
    const float* We2 = (const float*)d_in[23]; const float* be2 = (const float*)d_in[24];
    const float* Ws2 = (const float*)d_in[25]; const float* b2  = (const float*)d_in[26];

    const int N_ = in_sizes[3];        // batch_idx count == N
    const int E_ = in_sizes[2] / 2;    // edge_index is (2, E)
    const int* src = eidx;
    const int* dst = eidx + E_;

    // workspace carve-up (floats), 256B-aligned sections
    char* ws = (char*)d_ws;
    size_t off = 0;
    auto carve = [&](size_t bytes) {
        char* p = ws + off;
        off += (bytes + 255) & ~(size_t)255;
        return p;
    };
    float* eproj  = (float*)carve((size_t)E_ * 6 * sizeof(float));   // 38.4 MB
    float* kqv    = (float*)carve((size_t)N_ * 16 * sizeof(float));  //  6.4 MB
    float* agg1   = (float*)carve((size_t)N_ * H1 * sizeof(float));  //  2.0 MB
    float* hbuf   = (float*)carve((size_t)N_ * H1 * sizeof(float));  //  2.0 MB
    float* node3  = (float*)carve((size_t)N_ * 4 * sizeof(float));   //  1.6 MB
    float* agg2   = (float*)carve((size_t)N_ * sizeof(float));       //  0.4 MB
    float* Wcat   = (float*)carve((size_t)NDIM * 32 * sizeof(float));
    float* biasv  = (float*)carve(32 * sizeof(float));
    float* gsum   = (float*)carve(NGRP * H1 * sizeof(float));
    float* gcnt   = (float*)carve(NGRP * sizeof(float));
    float* mean   = (float*)carve(NGRP * H1 * sizeof(float));
    float* vsum   = (float*)carve(NGRP * H1 * sizeof(float));
    float* istd   = (float*)carve(NGRP * H1 * sizeof(float));

    const int TB = 256;
    const int ngE = (E_ + TB - 1) / TB;
    const int ngN = (N_ + TB - 1) / TB;
    const int mtiles = (N_ + 15) / 16;

    k_init<<<1, 512, 0, stream>>>(gsum, gcnt, vsum);
    k_pack<<<(NDIM * 32 + TB - 1) / TB, TB, 0, stream>>>(Wk1, Wq1, Wv1, Ws1,
                                                         bk1, bq1, bv1, b1,
                                                         Wcat, biasv);
    k_edge_proj<<<ngE, TB, 0, stream>>>(ea, We1, be1, We2, be2, eproj, E_);
    k_node_gemm<<<(mtiles + 7) / 8, TB, 0, stream>>>(x, Wcat, biasv, kqv, agg1, N_);
    k_edge_conv1<<<ngE, TB, 0, stream>>>(src, dst, kqv, eproj, agg1, E_);
    k_gn_sum<<<ngN, TB, 0, stream>>>(agg1, batch, gsum, gcnt, N_);
    k_gn_mean<<<1, NGRP * H1, 0, stream>>>(gsum, gcnt, mean);
    k_gn_center<<<ngN, TB, 0, stream>>>(agg1, batch, mean, gms, hbuf, vsum, N_);
    k_gn_istd<<<1, NGRP * H1, 0, stream>>>(vsum, gcnt, istd);
    k_gn_final<<<ngN, TB, 0, stream>>>(hbuf, batch, istd, gw, gb,
                                       Wk2, bk2, Wq2, bq2, Wv2, bv2, Ws2, b2,
                                       node3, agg2, N_);
    k_edge_conv2<<<ngE, TB, 0, stream>>>(src, dst, node3, eproj, agg2, E_);
    k_final<<<ngN, TB, 0, stream>>>(agg2, (float*)d_out, N_);
}